// YOLOLayer_17025250361539
// MI455X (gfx1250) — compile-verified
//
#include <hip/hip_runtime.h>
#include <cstdint>

#define GRID     76
#define NCH      85          // channels per anchor (4 box + 1 conf + 80 cls)
#define NANCH    3
#define CIN      (NANCH * NCH)   // 255
#define NPOS     (GRID * GRID)   // 5776
#define POS_TILE 152             // two grid rows per tile
#define NTILES   (NPOS / POS_TILE)  // 38
#define LDS_STRIDE 156           // padded row stride (floats): 156*4=624B, 16B aligned rows
#define THREADS  256

typedef float v4f __attribute__((ext_vector_type(4)));

// Low 32 bits of a generic pointer into __shared__ = LDS byte offset
// (flat LDS addressing truncates to addr[31:0], relative to wave LDS base).
__device__ __forceinline__ unsigned lds_off(const void* p) {
    return (unsigned)(uintptr_t)p;
}

__global__ __launch_bounds__(THREADS)
void yolo_head_kernel(const float* __restrict__ in, float* __restrict__ out) {
    __shared__ float smem[NCH * LDS_STRIDE];   // 53,040 bytes

    const int tile = blockIdx.x;   // 0..37  (which pair of grid rows)
    const int a    = blockIdx.y;   // anchor 0..2
    const int b    = blockIdx.z;   // batch

    const int tid  = threadIdx.x;
    const int pos0 = tile * POS_TILE;

    // ---------------- Phase 1: async global -> LDS (transpose staging) ----------------
    // Copy 85 rows x 152 floats as 85*38 = 3230 16-byte chunks.
    // Consecutive threads hit consecutive 16B chunks along pos => coalesced.
    const float* inBase = in + (size_t)(b * CIN + a * NCH) * NPOS + pos0;
    constexpr int CHUNKS_PER_ROW = POS_TILE / 4;      // 38
    constexpr int NCHUNKS = NCH * CHUNKS_PER_ROW;     // 3230

    for (int c = tid; c < NCHUNKS; c += THREADS) {
        int ch = c / CHUNKS_PER_ROW;
        int q  = c - ch * CHUNKS_PER_ROW;
        const float* gp = inBase + (size_t)ch * NPOS + (q << 2);
        unsigned la = lds_off(&smem[ch * LDS_STRIDE + (q << 2)]);
        asm volatile("global_load_async_to_lds_b128 %0, %1, off"
                     :: "v"(la), "v"((uint64_t)(uintptr_t)gp)
                     : "memory");
    }
    asm volatile("s_wait_asynccnt 0" ::: "memory");
    __syncthreads();

    // ---------------- Phase 2: transform + coalesced contiguous stores ----------------
    // Output block for this (b, a, tile) is a contiguous span of 152*85 floats.
    const float aw = (a == 0) ? 10.0f : (a == 1) ? 16.0f : 33.0f;  // exp(w)*anchor (stride cancels)
    const float ah = (a == 0) ? 13.0f : (a == 1) ? 30.0f : 23.0f;
    const int   gy0 = tile * 2;
    const float STRIDE = 8.0f;                    // 608 / 76
    const float NLOG2E = -1.44269504088896340736f;

    float* outBase = out + (size_t)((b * NANCH + a) * NPOS + pos0) * NCH;
    constexpr int NV4 = (POS_TILE * NCH) / 4;     // 3230 float4 stores

    for (int c = tid; c < NV4; c += THREADS) {
        v4f r;
        int flat = c << 2;                        // flat = p*85 + ch
        #pragma unroll
        for (int j = 0; j < 4; ++j) {
            int f  = flat + j;
            int p  = f / NCH;
            int ch = f - p * NCH;
            float v  = smem[ch * LDS_STRIDE + p];
            float e  = __builtin_amdgcn_exp2f(v * NLOG2E);     // exp(-v)
            float sg = __builtin_amdgcn_rcpf(1.0f + e);        // sigmoid(v)
            float rv;
            if (ch >= 4) {
                rv = sg;                                       // conf / class sigmoid
            } else if (ch == 0) {
                int gx = (p >= GRID) ? (p - GRID) : p;
                rv = (sg + (float)gx) * STRIDE;                // (sig(x)+grid_x)*stride
            } else if (ch == 1) {
                int gy = gy0 + ((p >= GRID) ? 1 : 0);
                rv = (sg + (float)gy) * STRIDE;                // (sig(y)+grid_y)*stride
            } else {
                float ex = __builtin_amdgcn_rcpf(e);           // exp(v)
                rv = ex * ((ch == 2) ? aw : ah);               // exp(wh)*ANCHOR
            }
            r[j] = rv;
        }
        __builtin_nontemporal_store(r, (v4f*)(outBase) + c);
    }
}

extern "C" void kernel_launch(void* const* d_in, const int* in_sizes, int n_in,
                              void* d_out, int out_size, void* d_ws, size_t ws_size,
                              hipStream_t stream) {
    (void)n_in; (void)d_ws; (void)ws_size; (void)out_size;
    const float* x = (const float*)d_in[0];
    float* out = (float*)d_out;
    const int B = in_sizes[0] / (CIN * NPOS);     // 32 for the reference shapes
    dim3 grid(NTILES, NANCH, B);
    yolo_head_kernel<<<grid, THREADS, 0, stream>>>(x, out);
}